// GatedDeltaRule_82764019794078
// MI455X (gfx1250) — compile-verified
//
#include <hip/hip_runtime.h>
#include <hip/hip_bf16.h>
#include <math.h>

// GatedDeltaRule for MI455X (gfx1250, wave32).
//   b=2, s=2048, h=1024, d=64, nh=16. M = b*s = 4096, K = 1024.
//
// Phase 1: q/k/v projections via V_WMMA_F32_16X16X4_F32, with double-buffered
//          LDS tiles filled by GLOBAL_LOAD_ASYNC_TO_LDS_B128 (ASYNCcnt) so the
//          async copy of K-block kb+1 overlaps the WMMAs on kb.
// Phase 2: g/beta sigmoid projections (tiny, N=16 each) -> scalar VALU.
// Phase 3: sequential scan (latency-bound): one workgroup per (b,head), thread
//          e owns state[*][e] in VGPRs; k/q/v of step t+1 are async-copied to
//          LDS while step t computes.

typedef __attribute__((ext_vector_type(2))) float v2f;
typedef __attribute__((ext_vector_type(8))) float v8f;

#define SEQ   2048
#define HDIM  1024
#define DHEAD 64
#define NHEAD 16
#define MTOT  4096            // b*s
#define NQKV  3072            // q|k|v concatenated
#define KTOT  1024
#define TILE_K 32
#define LDS_PITCH 36          // 32 + 4 pad: conflict-free b64 reads, 16B-aligned b128 stores

// Workspace layout (floats), (b,head,t,e) / (b,head,t) layouts for the scan:
#define WS_Q 0
#define WS_K (4096*1024)
#define WS_V (2*4096*1024)
#define WS_G (3*4096*1024)
#define WS_B (3*4096*1024 + 4096*16)

// generic LDS pointer -> 32-bit LDS byte address (low 32 bits of the shared
// aperture address) usable as the VDST operand of async-to-LDS instructions.
__device__ __forceinline__ unsigned int lds_addr(const void* p) {
    return (unsigned int)(unsigned long long)p;
}

__device__ __forceinline__ void async_load_b128(unsigned int lds, const void* gptr) {
    asm volatile("global_load_async_to_lds_b128 %0, %1, off"
                 :: "v"(lds), "v"((unsigned long long)gptr) : "memory");
}
__device__ __forceinline__ void async_load_b32(unsigned int lds, const void* gptr) {
    asm volatile("global_load_async_to_lds_b32 %0, %1, off"
                 :: "v"(lds), "v"((unsigned long long)gptr) : "memory");
}

// ---------------------------------------------------------------------------
// Phase 1: C[m, n] = sum_k hs[m,k] * W[n,k]   (n in 0..3071 selects Wq/Wk/Wv)
// Block: 128 threads = 4 waves. Block tile 64(M) x 64(N). K-blocked by 32,
// double-buffered via async-to-LDS. Each wave computes a 16x64 strip.
// ---------------------------------------------------------------------------
__global__ __launch_bounds__(128) void qkv_gemm_wmma(
    const float* __restrict__ hs, const float* __restrict__ Wq,
    const float* __restrict__ Wk, const float* __restrict__ Wv,
    float* __restrict__ ws)
{
    __shared__ float As[2][64 * LDS_PITCH];   // A tile: 64 rows (m) x 32 (k)
    __shared__ float Bs[2][64 * LDS_PITCH];   // W tile: 64 rows (n) x 32 (k)

    const int nblk   = blockIdx.x;              // 0..47
    const int mblk   = blockIdx.y;              // 0..63
    const int n_base = nblk * 64;               // 0..3008
    const int which  = n_base >> 10;            // 0=q,1=k,2=v (tile never straddles)
    const int n_in   = n_base & 1023;           // column within that weight
    const float* W   = (which == 0) ? Wq : ((which == 1) ? Wk : Wv);
    const int m0     = mblk * 64;

    const int tid  = threadIdx.x;
    const int lane = tid & 31;
    const int wave = tid >> 5;                  // 0..3
    const int lr   = tid >> 3;                  // 0..15  (staging row group)
    const int lc   = tid & 7;                   // 0..7   (staging float4 col)

    v8f acc[4];
    #pragma unroll
    for (int j = 0; j < 4; ++j)
        #pragma unroll
        for (int r = 0; r < 8; ++r) acc[j][r] = 0.0f;

    const int mrow  = wave * 16 + (lane & 15);  // A row this lane feeds
    const int khalf = (lane >> 4) * 2;          // lanes 16-31 hold K+2,K+3

    // issue one K-tile's async copies: 8 b128 instructions per wave per tile
    auto issue_tile = [&](int buf, int kb) {
        #pragma unroll
        for (int i = 0; i < 4; ++i) {
            const int row = lr + 16 * i;
            async_load_b128(lds_addr(&As[buf][row * LDS_PITCH + lc * 4]),
                            &hs[(size_t)(m0 + row) * KTOT + kb + lc * 4]);
            async_load_b128(lds_addr(&Bs[buf][row * LDS_PITCH + lc * 4]),
                            &W[(size_t)(n_in + row) * KTOT + kb + lc * 4]);
        }
    };

    issue_tile(0, 0);

    for (int kb = 0; kb < KTOT; kb += TILE_K) {
        const int cur = (kb >> 5) & 1;
        if (kb + TILE_K < KTOT) {
            issue_tile(cur ^ 1, kb + TILE_K);          // prefetch next tile
            asm volatile("s_wait_asynccnt 0x8" ::: "memory");  // prev tile done
        } else {
            asm volatile("s_wait_asynccnt 0x0" ::: "memory");
        }
        __syncthreads();   // all waves' copies for `cur` visible

        // --- 8 k-steps of 4, 4 WMMAs each ---
        #pragma unroll
        for (int kk = 0; kk < TILE_K; kk += 4) {
            v2f av = *reinterpret_cast<const v2f*>(
                &As[cur][mrow * LDS_PITCH + kk + khalf]);
            #pragma unroll
            for (int j = 0; j < 4; ++j) {
                v2f bv = *reinterpret_cast<const v2f*>(
                    &Bs[cur][(j * 16 + (lane & 15)) * LDS_PITCH + kk + khalf]);
                acc[j] = __builtin_amdgcn_wmma_f32_16x16x4_f32(
                    /*neg_a=*/false, av, /*neg_b=*/false, bv,
                    /*c_mod=*/(short)0, acc[j], /*reuse_a=*/false, /*reuse_b=*/false);
            }
        }
        __syncthreads();   // done reading `cur` before it is refilled at kb+2
    }

    // --- epilogue: scatter into (b, head, t, e) layout for the scan ---
    float* dst = ws + (size_t)which * (size_t)(4096 * 1024);
    #pragma unroll
    for (int j = 0; j < 4; ++j) {
        const int gn   = n_in + j * 16 + (lane & 15);   // channel 0..1023
        const int head = gn >> 6;
        const int e    = gn & 63;
        #pragma unroll
        for (int r = 0; r < 8; ++r) {
            const int gm = m0 + wave * 16 + r + ((lane >> 4) << 3);
            const int bb = gm >> 11;          // batch
            const int t  = gm & 2047;         // time
            dst[(((size_t)bb * NHEAD + head) * SEQ + t) * DHEAD + e] = acc[j][r];
        }
    }
}

// ---------------------------------------------------------------------------
// Phase 2: g = sigmoid(hs @ Wg.T), beta = sigmoid(hs @ Wb.T).  One dot/thread.
// ---------------------------------------------------------------------------
__global__ __launch_bounds__(256) void gb_kernel(
    const float* __restrict__ hs, const float* __restrict__ Wg,
    const float* __restrict__ Wb, float* __restrict__ gout,
    float* __restrict__ bout)
{
    const int gid = blockIdx.x * 256 + threadIdx.x;   // 0 .. 4096*32-1
    const int m = gid >> 5;
    const int o = gid & 31;                           // 0-15: g rows, 16-31: beta rows
    const float* wrow = (o < 16) ? (Wg + (size_t)o * KTOT)
                                 : (Wb + (size_t)(o - 16) * KTOT);
    const float* hrow = hs + (size_t)m * KTOT;

    float acc = 0.0f;
    for (int kk = 0; kk < KTOT; kk += 4) {
        float4 hv = *reinterpret_cast<const float4*>(&hrow[kk]);
        float4 wv = *reinterpret_cast<const float4*>(&wrow[kk]);
        acc = fmaf(hv.x, wv.x, acc);
        acc = fmaf(hv.y, wv.y, acc);
        acc = fmaf(hv.z, wv.z, acc);
        acc = fmaf(hv.w, wv.w, acc);
    }
    const float sig = 1.0f / (1.0f + __expf(-acc));
    const int bb = m >> 11, t = m & 2047, head = o & 15;
    float* dst = (o < 16) ? gout : bout;
    dst[((size_t)bb * NHEAD + head) * SEQ + t] = sig;
}

// ---------------------------------------------------------------------------
// Phase 3: sequential gated-delta scan. Grid = 32 (b*nh), block = 64 threads.
// Thread e owns state[0..63][e] in registers; per-step k/q/v staged to LDS by
// async-to-LDS copies issued one step ahead (hides HBM/L2 latency on the
// serial dependence chain).
// ---------------------------------------------------------------------------
__global__ __launch_bounds__(64) void scan_kernel(
    const float* __restrict__ q, const float* __restrict__ k,
    const float* __restrict__ v, const float* __restrict__ g,
    const float* __restrict__ beta, float* __restrict__ out,
    float* __restrict__ state_out)
{
    const int bh = blockIdx.x;        // b*16 + head
    const int e  = threadIdx.x;       // 0..63 value-dim column
    const int bb = bh >> 4;
    const int head = bh & 15;

    const float* qh = q + (size_t)bh * SEQ * DHEAD;
    const float* kh = k + (size_t)bh * SEQ * DHEAD;
    const float* vh = v + (size_t)bh * SEQ * DHEAD;
    const float* gh = g + (size_t)bh * SEQ;
    const float* bth = beta + (size_t)bh * SEQ;

    float st[DHEAD];
    #pragma unroll
    for (int d2 = 0; d2 < DHEAD; ++d2) st[d2] = 0.0f;

    __shared__ float sk[2][DHEAD], sq[2][DHEAD], sv[2][DHEAD];

    // issue step-t staging: 3 async b32 instructions per wave
    auto issue_step = [&](int buf, int t) {
        async_load_b32(lds_addr(&sk[buf][e]), &kh[(size_t)t * DHEAD + e]);
        async_load_b32(lds_addr(&sq[buf][e]), &qh[(size_t)t * DHEAD + e]);
        async_load_b32(lds_addr(&sv[buf][e]), &vh[(size_t)t * DHEAD + e]);
    };

    issue_step(0, 0);

    for (int t = 0; t < SEQ; ++t) {
        const int cur = t & 1;
        if (t + 1 < SEQ) {
            issue_step(cur ^ 1, t + 1);                 // stage next step
            asm volatile("s_wait_asynccnt 0x3" ::: "memory");  // step t done
        } else {
            asm volatile("s_wait_asynccnt 0x0" ::: "memory");
        }
        __syncthreads();

        const float gt = gh[t];        // wave-uniform -> scalar load
        const float bt = bth[t];

        float kreg[DHEAD];
        float vp = 0.0f;
        #pragma unroll
        for (int d2 = 0; d2 < DHEAD; ++d2) {
            const float kd = sk[cur][d2];
            kreg[d2] = kd;
            st[d2] *= gt;                       // gated decay
            vp = fmaf(kd, st[d2], vp);          // v_pred = k @ state
        }
        const float delta = (sv[cur][e] - vp) * bt;
        float o = 0.0f;
        #pragma unroll
        for (int d2 = 0; d2 < DHEAD; ++d2) {
            st[d2] = fmaf(kreg[d2], delta, st[d2]);   // rank-1 update
            o = fmaf(sq[cur][d2], st[d2], o);         // out = q @ state
        }
        out[((size_t)bb * SEQ + t) * HDIM + head * DHEAD + e] = o;
        __syncthreads();   // done reading `cur` before it is refilled at t+2
    }

    #pragma unroll
    for (int d2 = 0; d2 < DHEAD; ++d2)
        state_out[((size_t)bh * DHEAD + d2) * DHEAD + e] = st[d2];
}

// ---------------------------------------------------------------------------
extern "C" void kernel_launch(void* const* d_in, const int* in_sizes, int n_in,
                              void* d_out, int out_size, void* d_ws, size_t ws_size,
                              hipStream_t stream)
{
    const float* hs = (const float*)d_in[0];
    const float* Wq = (const float*)d_in[1];
    const float* Wk = (const float*)d_in[2];
    const float* Wv = (const float*)d_in[3];
    const float* Wg = (const float*)d_in[4];
    const float* Wb = (const float*)d_in[5];

    float* ws  = (float*)d_ws;
    float* qws = ws + WS_Q;
    float* kws = ws + WS_K;
    float* vws = ws + WS_V;
    float* gws = ws + WS_G;
    float* bws = ws + WS_B;

    float* out       = (float*)d_out;                     // (b, s, h)
    float* state_out = out + (size_t)MTOT * HDIM;         // (b, nh, d, d)

    qkv_gemm_wmma<<<dim3(NQKV / 64, MTOT / 64), 128, 0, stream>>>(hs, Wq, Wk, Wv, ws);
    gb_kernel<<<(MTOT * 32) / 256, 256, 0, stream>>>(hs, Wg, Wb, gws, bws);
    scan_kernel<<<2 * NHEAD, 64, 0, stream>>>(qws, kws, vws, gws, bws, out, state_out);
}